// GCNConv_11347303596492
// MI455X (gfx1250) — compile-verified
//
#include <hip/hip_runtime.h>

#define FEAT  128
#define EMBED 128

typedef float v2f __attribute__((ext_vector_type(2)));
typedef float v8f __attribute__((ext_vector_type(8)));

// ---------------- zero-fill ints (deg + cursor; harness poisons ws with 0xAA)
__global__ void zero_int_kernel(int* __restrict__ p, int n) {
    int i = blockIdx.x * blockDim.x + threadIdx.x;
    int stride = gridDim.x * blockDim.x;
    for (; i < n; i += stride) p[i] = 0;
}

// ---------------- deg[i] = # edges with row==i  (L2-resident int atomics, E ops only)
__global__ void deg_kernel(const int* __restrict__ row, int* __restrict__ deg, int E) {
    int e = blockIdx.x * blockDim.x + threadIdx.x;
    if (e < E) atomicAdd(&deg[row[e]], 1);
}

// ---------------- exclusive prefix sum over deg[N] -> rowstart[N+1], single block.
// 1024 threads: serial chunk sums, LDS Hillis-Steele scan of thread sums, write-back.
__global__ __launch_bounds__(1024) void scan_kernel(const int* __restrict__ deg,
                                                    int* __restrict__ rowstart,
                                                    int N, int E) {
    __shared__ int sums[1024];
    const int t = threadIdx.x;
    const int chunk = (N + 1023) / 1024;
    const int begin = t * chunk;
    const int end = (begin + chunk < N) ? (begin + chunk) : N;

    int s = 0;
    for (int i = begin; i < end; ++i) s += deg[i];
    sums[t] = s;
    __syncthreads();

    // inclusive scan over the 1024 thread sums
    for (int off = 1; off < 1024; off <<= 1) {
        int v = (t >= off) ? sums[t - off] : 0;
        __syncthreads();
        sums[t] += v;
        __syncthreads();
    }

    int run = (t == 0) ? 0 : sums[t - 1];   // exclusive offset of this chunk
    for (int i = begin; i < end; ++i) {
        rowstart[i] = run;
        run += deg[i];
    }
    if (t == 0) rowstart[N] = E;
}

// ---------------- bin edges into CSR: csr_col[rowstart[r] + cursor[r]++] = col[e]
__global__ void csr_kernel(const int* __restrict__ row, const int* __restrict__ col,
                           const int* __restrict__ rowstart, int* __restrict__ cursor,
                           int* __restrict__ csr_col, int E) {
    int e = blockIdx.x * blockDim.x + threadIdx.x;
    if (e < E) {
        const int r = row[e];
        const int pos = atomicAdd(&cursor[r], 1);
        csr_col[rowstart[r] + pos] = col[e];
    }
}

// ---------------- h = x @ W via V_WMMA_F32_16X16X4_F32
// block = 256 threads = 8 waves; block computes 16 rows x 128 cols.
// wave w handles the 16x16 tile at cols n0 = 16*w; K loop = 32 steps of 4.
__global__ __launch_bounds__(256) void gemm_wmma_kernel(const float* __restrict__ x,
                                                        const float* __restrict__ W,
                                                        float* __restrict__ h,
                                                        int nrows) {
    const int lane = threadIdx.x & 31;
    const int wave = threadIdx.x >> 5;
    const int l  = lane & 15;   // M (for A) / N (for B,C,D)
    const int hh = lane >> 4;   // selects K pair (A/B) and M half (C/D)
    const int m0 = blockIdx.x * 16;
    const int n0 = wave * 16;

    // clamp row (keeps EXEC all-ones through the WMMA loop for a partial tail tile)
    int mrow = m0 + l;
    if (mrow >= nrows) mrow = nrows - 1;
    const float* __restrict__ xrow = x + (size_t)mrow * FEAT;

    v8f acc = {};
#pragma unroll 8
    for (int k0 = 0; k0 < FEAT; k0 += 4) {
        const int ka = k0 + 2 * hh;
        v2f a, b;
        // A (16x4): lane l holds row m0+l, K = ka, ka+1  (contiguous -> b64 load)
        a.x = xrow[ka];
        a.y = xrow[ka + 1];
        // B (4x16): lane l holds col n0+l, K = ka, ka+1
        b.x = W[(size_t)ka       * EMBED + n0 + l];
        b.y = W[(size_t)(ka + 1) * EMBED + n0 + l];
        // (neg_a, A, neg_b, B, c_mod, C, reuse_a, reuse_b)
        acc = __builtin_amdgcn_wmma_f32_16x16x4_f32(false, a, false, b,
                                                    (short)0, acc, false, false);
    }

    // C/D layout: acc[r] = C[m0 + r + 8*hh][n0 + l]
    float* __restrict__ hout = h + (size_t)m0 * EMBED + n0 + l;
    if (m0 + 16 <= nrows) {          // uniform branch -> straight-line stores
#pragma unroll
        for (int r = 0; r < 8; ++r)
            hout[(size_t)(r + 8 * hh) * EMBED] = acc[r];
    } else {
#pragma unroll
        for (int r = 0; r < 8; ++r) {
            const int mr = r + 8 * hh;
            if (m0 + mr < nrows) hout[(size_t)mr * EMBED] = acc[r];
        }
    }
}

// ---------------- pull-mode aggregation: one wave per node, zero atomics.
// lane covers 4 of the 128 features; inner loop is one b128 L2-resident gather
// + 4 FMAs per neighbor; out written exactly once with b128 stores.
__global__ __launch_bounds__(256) void gather_kernel(const float* __restrict__ h,
                                                     const int* __restrict__ deg,
                                                     const int* __restrict__ rowstart,
                                                     const int* __restrict__ csr_col,
                                                     float* __restrict__ out, int N) {
    const int lane = threadIdx.x & 31;
    const int r = blockIdx.x * (blockDim.x >> 5) + (threadIdx.x >> 5);
    if (r >= N) return;

    const int start = rowstart[r];
    const int end   = rowstart[r + 1];
    const float dr  = (float)deg[r];

    float4 acc = {0.f, 0.f, 0.f, 0.f};
    for (int i = start; i < end; ++i) {
        const int c = csr_col[i];
        const float norm = rsqrtf(dr * (float)deg[c]);
        const float4 v = ((const float4*)(h + (size_t)c * EMBED))[lane];
        acc.x += v.x * norm;
        acc.y += v.y * norm;
        acc.z += v.z * norm;
        acc.w += v.w * norm;
    }
    ((float4*)(out + (size_t)r * EMBED))[lane] = acc;
}

extern "C" void kernel_launch(void* const* d_in, const int* in_sizes, int n_in,
                              void* d_out, int out_size, void* d_ws, size_t ws_size,
                              hipStream_t stream) {
    const float* x   = (const float*)d_in[0];   // [N, 128]
    const float* W   = (const float*)d_in[1];   // [128, 128]
    const int*   row = (const int*)d_in[2];     // [E]
    const int*   col = (const int*)d_in[3];     // [E]
    float*       out = (float*)d_out;           // [N, 128]

    const int N = in_sizes[0] / FEAT;
    const int E = in_sizes[2];

    // workspace layout (16B-aligned head):
    //   h        float[N*128]
    //   deg      int[N]
    //   cursor   int[N]      (contiguous with deg -> one zero-fill)
    //   rowstart int[N+1]
    //   csr_col  int[E]
    float* h        = (float*)d_ws;
    int*   deg      = (int*)(h + (size_t)N * EMBED);
    int*   cursor   = deg + N;
    int*   rowstart = cursor + N;
    int*   csr_col  = rowstart + (N + 1);

    zero_int_kernel<<<256, 256, 0, stream>>>(deg, 2 * N);            // deg + cursor

    deg_kernel<<<(E + 255) / 256, 256, 0, stream>>>(row, deg, E);

    scan_kernel<<<1, 1024, 0, stream>>>(deg, rowstart, N, E);

    csr_kernel<<<(E + 255) / 256, 256, 0, stream>>>(row, col, rowstart, cursor, csr_col, E);

    gemm_wmma_kernel<<<(N + 15) / 16, 256, 0, stream>>>(x, W, h, N);

    gather_kernel<<<(N + 7) / 8, 256, 0, stream>>>(h, deg, rowstart, csr_col, out, N);
}